// RosenfeldErkModel_77360950936300
// MI455X (gfx1250) — compile-verified
//
#include <hip/hip_runtime.h>
#include <math.h>

#ifndef __has_builtin
#define __has_builtin(x) 0
#endif

typedef __attribute__((ext_vector_type(2))) float        v2f;
typedef __attribute__((ext_vector_type(8))) float        v8f;
typedef __attribute__((ext_vector_type(4))) unsigned int v4u;
typedef __attribute__((ext_vector_type(4))) int          v4i;
typedef __attribute__((ext_vector_type(8))) int          v8i;

#define D        256
#define D3       768
#define BATCH    128
#define NSAMP    7                 // 1 pos_u + 1 pos_v + 5 neg
#define ROWS     (BATCH * NSAMP)   // 896
#define KB_BLOCKS 12               // k split: 12 blocks of 64
#define ISUB     8                 // i-planes per staged tile (double-buffered)
#define NSTAGE   (D / ISUB)        // 32 stages per k-chunk
#define TLDS_ROW 264               // 16j*16k + 8 pad floats per i-plane (bank-conflict free)
#define TLDS_FLOATS (ISUB * TLDS_ROW)  // 2112 floats per buffer
#define M2_BSTR  272               // per-b M tile: 16j * (16k + 1 pad)
#define SMEM_CONTRACT ((2 * TLDS_FLOATS + BATCH * M2_BSTR) * 4)  // 156,160 B < 160 KB

// ---------------- WMMA f32 16x16x4 (full-precision matrix core) ----------------
#if defined(__gfx1250__) && __has_builtin(__builtin_amdgcn_wmma_f32_16x16x4_f32)
#define HAVE_WMMA_F32 1
#else
#define HAVE_WMMA_F32 0
#endif

__device__ __forceinline__ v8f wmma4(v2f a, v2f b, v8f c) {
#if HAVE_WMMA_F32
  // 8-arg pattern per probe: (neg_a, A, neg_b, B, c_mod, C, reuse_a, reuse_b)
  return __builtin_amdgcn_wmma_f32_16x16x4_f32(false, a, false, b, (short)0, c,
                                               false, false);
#else
  // placeholder so non-gfx1250 / host parse compiles
  c[0] += a.x * b.x + a.y * b.y;
  return c;
#endif
}

// ---------------- Tensor Data Mover staging: T tile (16k x 16j x ISUBi) -> LDS --
#if defined(__gfx1250__)
#define USE_TDM 1
#else
#define USE_TDM 0
#endif

#if USE_TDM
__device__ __forceinline__ void wait_tensorcnt0() {
#if __has_builtin(__builtin_amdgcn_s_wait_tensorcnt)
  __builtin_amdgcn_s_wait_tensorcnt((short)0);
#else
  asm volatile("s_wait_tensorcnt 0x0" ::: "memory");
#endif
}

__device__ __forceinline__ void tdm_tile_load(const float* gsrc, unsigned lds_off) {
  // D#: tile dim0=k(16), dim1=j(16), dim2=i(ISUB); strides 768 / 196608 elems;
  // pad_enable: 8 dwords every 256 dwords => LDS i-plane stride = 264 floats.
  unsigned long long ga = (unsigned long long)(uintptr_t)gsrc;
  v4u g0;
  g0[0] = 1u;                                   // count=1, user mode
  g0[1] = lds_off;                              // lds_addr (bytes)
  g0[2] = (unsigned)(ga & 0xFFFFFFFFu);         // global_addr lo
  g0[3] = (unsigned)((ga >> 32) & 0x01FFFFFFu) | 0x80000000u;  // addr hi | type=2
  v8i g1;
  g1[0] = (int)((2u << 16) | (1u << 20) | (7u << 22) | (7u << 25)); // 4B, pad 8dw/256dw
  g1[1] = (int)(16u << 16);                     // tensor_dim0 = 16 (lo16)
  g1[2] = (int)(16u << 16);                     // tensor_dim1 = 16 (lo16)
  g1[3] = (int)(16u << 16);                     // tile_dim0 = 16
  g1[4] = (int)(16u | ((unsigned)ISUB << 16));  // tile_dim1 = 16, tile_dim2 = ISUB
  g1[5] = 768;                                  // tensor_dim0_stride lo32
  g1[6] = 0;                                    // stride0 hi16 | stride1 lo16 (=0)
  g1[7] = 3;                                    // stride1 >> 16 (196608 = 3<<16)
  v4i g2; g2[0] = ISUB; g2[1] = 0; g2[2] = 0; g2[3] = 0;  // tensor_dim2 = ISUB
  v4i g3; g3[0] = 0;  g3[1] = 0; g3[2] = 0; g3[3] = 0;
#if __has_builtin(__builtin_amdgcn_tensor_load_to_lds)
#if defined(__clang_major__) && __clang_major__ >= 23
  v8i g4; g4[0]=0; g4[1]=0; g4[2]=0; g4[3]=0; g4[4]=0; g4[5]=0; g4[6]=0; g4[7]=0;
  __builtin_amdgcn_tensor_load_to_lds(g0, g1, g2, g3, g4, 0);
#else
  __builtin_amdgcn_tensor_load_to_lds(g0, g1, g2, g3, 0);
#endif
#else
  // Direct CDNA5 ISA: VIMAGE tensor op, descriptor groups in SGPR tuples.
  asm volatile("tensor_load_to_lds %0, %1, %2, %3"
               :
               : "s"(g0), "s"(g1), "s"(g2), "s"(g3)
               : "memory");
#endif
}
#endif  // USE_TDM

// ======================= Kernel A: tv / tvB (small dense layers) ================
__global__ __launch_bounds__(256) void k_time(
    const float* __restrict__ time, const float* __restrict__ m1,
    const float* __restrict__ b1, const float* __restrict__ W2,
    const float* __restrict__ b2, const float* __restrict__ Bm,
    float* __restrict__ tv, float* __restrict__ tvB) {
  const int b = blockIdx.x, j = threadIdx.x;
  __shared__ float h1s[D], tvs[D];
  const float t = time[b] * (1.0f / 20.0f);
  h1s[j] = tanhf(t * m1[j] + b1[j]);
  __syncthreads();
  float acc = b2[j];
  const float* w2r = W2 + (size_t)j * D;
#pragma unroll 8
  for (int i = 0; i < D; ++i) acc += h1s[i] * w2r[i];
  const float tvj = tanhf(acc);
  tvs[j] = tvj;
  tv[(size_t)b * D + j] = tvj;
  __syncthreads();
  float accB = 0.f;
#pragma unroll 8
  for (int i = 0; i < D; ++i) accB += tvs[i] * Bm[(size_t)i * D + j];
  tvB[(size_t)b * D + j] = accB;
}

// ======================= Kernel G: gather embeddings E[b][s][k] =================
__global__ __launch_bounds__(256) void k_gather(
    const int* __restrict__ pos_u, const int* __restrict__ pos_v,
    const int* __restrict__ neg_v, const float* __restrict__ U,
    const float* __restrict__ V, float* __restrict__ E) {
  const int b = blockIdx.x, s = blockIdx.y;
  const float* src;
  if (s == 0)      src = U + (size_t)pos_u[b] * D3;
  else if (s == 1) src = V + (size_t)pos_v[b] * D3;
  else             src = V + (size_t)neg_v[b * 5 + (s - 2)] * D3;
  float* dst = E + ((size_t)b * NSAMP + s) * D3;
  for (int i = threadIdx.x; i < D3; i += 256) dst[i] = src[i];
}

// ========== Kernel B: fused  M = tv x T  and  h3 += E x M  (T read ONCE) ========
// grid (16 j-tiles, 12 k-blocks); 256 thr = 8 waves; wave w owns b-tile w (16 b's)
// T staging is TDM double-buffered: DMA for stage n+1 overlaps compute of stage n.
__global__ __launch_bounds__(256) void k_contract(
    const float* __restrict__ Tg, const float* __restrict__ tvg,
    const float* __restrict__ Eg, float* __restrict__ h3p) {
  const int tid = threadIdx.x;
  const int w = tid >> 5, lane = tid & 31;
  const int bl = lane & 15, hi = lane >> 4;
  const int j0 = blockIdx.x << 4;
  const int kb = blockIdx.y;
  extern __shared__ float smem[];
  float* tbuf0 = smem;                        // ISUB * 264 floats
  float* tbuf1 = smem + TLDS_FLOATS;          // ISUB * 264 floats
  float* m2    = smem + 2 * TLDS_FLOATS;      // 128 * 272 floats
  const int brow = (w << 4) + bl;             // A rows of this wave's b-tile

  const v8f vzero = {0.f, 0.f, 0.f, 0.f, 0.f, 0.f, 0.f, 0.f};
  v8f acc2[16];                               // h3 accumulators: 16 b x (16s x 16j)
#pragma unroll
  for (int i = 0; i < 16; ++i) acc2[i] = vzero;

  for (int c = 0; c < 4; ++c) {               // 4 chunks of 16 k in this k-block
    const int k0 = (kb << 6) + (c << 4);
    const float* gchunk = Tg + (size_t)j0 * D3 + k0;
    v8f acc1[16];                             // M tiles: 16 j x (16b x 16k)
#pragma unroll
    for (int t = 0; t < 16; ++t) acc1[t] = vzero;

    // ---- prologue: stage 0 into buffer 0 ----
#if USE_TDM
    if (tid < 32) {
      tdm_tile_load(gchunk, (unsigned)(uintptr_t)(void*)tbuf0);
      wait_tensorcnt0();
    }
#else
    for (int e = tid; e < ISUB * 256; e += 256) {
      const int il = e >> 8, r = e & 255, jl = r >> 4, kl = r & 15;
      tbuf0[il * TLDS_ROW + jl * 16 + kl] =
          gchunk[(size_t)il * (D * D3) + (size_t)jl * D3 + kl];
    }
#endif
    __syncthreads();

    for (int st = 0; st < NSTAGE; ++st) {
      float* cur = (st & 1) ? tbuf1 : tbuf0;
      float* nxt = (st & 1) ? tbuf0 : tbuf1;
      // ---- kick off DMA for stage st+1 (overlaps compute below) ----
      if (st + 1 < NSTAGE) {
        const float* gsrc = gchunk + (size_t)(st + 1) * ISUB * (D * D3);
#if USE_TDM
        if (tid < 32) tdm_tile_load(gsrc, (unsigned)(uintptr_t)(void*)nxt);
#else
        for (int e = tid; e < ISUB * 256; e += 256) {
          const int il = e >> 8, r = e & 255, jl = r >> 4, kl = r & 15;
          nxt[il * TLDS_ROW + jl * 16 + kl] =
              gsrc[(size_t)il * (D * D3) + (size_t)jl * D3 + kl];
        }
#endif
      }
      // ---- GEMM1 over the ISUB staged i-planes ----
      const int i0 = st * ISUB;
#pragma unroll
      for (int is = 0; is < ISUB / 4; ++is) {
        const int gi = i0 + (is << 2);
        v2f a;
        a.x = tvg[(size_t)brow * D + gi + 2 * hi];
        a.y = tvg[(size_t)brow * D + gi + 2 * hi + 1];
#pragma unroll
        for (int t = 0; t < 16; ++t) {        // 16 j columns, one WMMA tile each
          const int rb = (is * 4 + 2 * hi) * TLDS_ROW + t * 16 + bl;
          v2f bf;
          bf.x = cur[rb];
          bf.y = cur[rb + TLDS_ROW];
          acc1[t] = wmma4(a, bf, acc1[t]);
        }
      }
#if USE_TDM
      if (tid < 32) wait_tensorcnt0();        // stage st+1 landed
#endif
      __syncthreads();
    }

    // ---- spill M tiles to LDS as m2[b][j][k(+pad)] ----
#pragma unroll
    for (int t = 0; t < 16; ++t) {
      const int bbase = (w << 4) + 8 * hi;
#pragma unroll
      for (int r = 0; r < 8; ++r)
        m2[(bbase + r) * M2_BSTR + t * 17 + bl] = acc1[t][r];
    }
    __syncthreads();

    // ---- GEMM2: per-b (16s x 16k) @ (16k x 16j); s rows 7..15 are don't-care ---
    for (int bb = 0; bb < 16; ++bb) {
      const int b = (w << 4) + bb;
      const int srow = (bl < 7) ? bl : 0;
      const float* erow = Eg + ((size_t)b * NSAMP + srow) * D3 + k0;
      v2f ea[4];
#pragma unroll
      for (int ks = 0; ks < 4; ++ks) {        // prefetch all A fragments first
        ea[ks].x = erow[ks * 4 + 2 * hi];
        ea[ks].y = erow[ks * 4 + 2 * hi + 1];
      }
#pragma unroll
      for (int ks = 0; ks < 4; ++ks) {
        v2f bf;
        bf.x = m2[b * M2_BSTR + bl * 17 + ks * 4 + 2 * hi];
        bf.y = m2[b * M2_BSTR + bl * 17 + ks * 4 + 2 * hi + 1];
        acc2[bb] = wmma4(ea[ks], bf, acc2[bb]);
      }
    }
    __syncthreads();
  }

  // ---- write h3 partials for this k-block (deterministic: no atomics) ----
#pragma unroll
  for (int bb = 0; bb < 16; ++bb) {
    const int b = (w << 4) + bb;
#pragma unroll
    for (int r = 0; r < 8; ++r) {
      const int s = r + 8 * hi;
      if (s < 7)
        h3p[(((size_t)kb * BATCH + b) * NSAMP + s) * D + j0 + bl] = acc2[bb][r];
    }
  }
}

// ======================= Kernel C0: transpose W4 -> W4t[i][o] ===================
__global__ __launch_bounds__(256) void k_transpose(const float* __restrict__ W4,
                                                   float* __restrict__ W4t) {
  const int o = blockIdx.x, i = threadIdx.x;  // grid 768, block 256
  W4t[(size_t)i * D3 + o] = W4[(size_t)o * D + i];
}

// ============ Kernel C1: reduce k-block partials + tvB broadcast ================
__global__ __launch_bounds__(256) void k_reduce(const float* __restrict__ h3p,
                                                const float* __restrict__ tvB,
                                                float* __restrict__ h3) {
  const int row = blockIdx.x;  // 896 = b*7+s
  const int j = threadIdx.x;
  const int b = row / NSAMP;
  float acc = tvB[(size_t)b * D + j];
#pragma unroll
  for (int p = 0; p < KB_BLOCKS; ++p)
    acc += h3p[((size_t)p * ROWS + row) * D + j];
  h3[(size_t)row * D + j] = acc;
}

// ======== Kernel C2: use = h3 @ W4^T + b4  (WMMA, coalesced via W4t) ============
__global__ __launch_bounds__(256) void k_dense4(const float* __restrict__ h3,
                                                const float* __restrict__ W4t,
                                                const float* __restrict__ b4,
                                                float* __restrict__ use_) {
  const int w = threadIdx.x >> 5, lane = threadIdx.x & 31;
  const int bl = lane & 15, hi = lane >> 4;
  const int rt = blockIdx.x;                // 56 row tiles (rows 896)
  const int ct = blockIdx.y * 8 + w;        // 48 col tiles (cols 768)
  const int o0 = ct << 4;
  const float c0 = b4[o0 + bl];
  v8f acc = {c0, c0, c0, c0, c0, c0, c0, c0};
  const float* arow = h3 + (size_t)((rt << 4) + bl) * D + 2 * hi;
  const float* bcol = W4t + (size_t)(2 * hi) * D3 + o0 + bl;
#pragma unroll 4
  for (int ks = 0; ks < 64; ++ks) {
    v2f a, bf;
    a.x = arow[ks * 4];
    a.y = arow[ks * 4 + 1];
    bf.x = bcol[(size_t)(ks * 4) * D3];
    bf.y = bcol[(size_t)(ks * 4 + 1) * D3];
    acc = wmma4(a, bf, acc);
  }
#pragma unroll
  for (int r = 0; r < 8; ++r)
    use_[(size_t)((rt << 4) + r + 8 * hi) * D3 + o0 + bl] = acc[r];
}

// ============ Kernel D: per-b dots, clip, -log_sigmoid  (wave32 reduce) =========
__device__ __forceinline__ float softplusf(float t) {  // log(1+e^t), stable
  return (t > 0.f) ? (t + log1pf(expf(-t))) : log1pf(expf(t));
}

__global__ __launch_bounds__(192) void k_score(const float* __restrict__ use_,
                                               float* __restrict__ scoreb,
                                               float* __restrict__ negpart) {
  const int b = blockIdx.x;
  const int w = threadIdx.x >> 5, lane = threadIdx.x & 31;  // 6 waves
  __shared__ float uw[D3];
  for (int i = threadIdx.x; i < D3; i += 192)
    uw[i] = use_[(size_t)(b * NSAMP) * D3 + i];
  __syncthreads();
  const float* cr = use_ + (size_t)(b * NSAMP + 1 + w) * D3;
  float p = 0.f;
  for (int i = lane; i < D3; i += 32) p += uw[i] * cr[i];
#pragma unroll
  for (int off = 16; off; off >>= 1) p += __shfl_down(p, off);
  if (lane == 0) {
    const float x = fminf(10.f, fmaxf(-10.f, p));
    if (w == 0) scoreb[b] = softplusf(-x);               // -log_sigmoid(score)
    else        negpart[b * 5 + (w - 1)] = softplusf(x); // -log_sigmoid(-neg)
  }
}

// ======================= Kernel E: means -> 3 outputs ===========================
__global__ __launch_bounds__(128) void k_final(const float* __restrict__ scoreb,
                                               const float* __restrict__ negpart,
                                               float* __restrict__ out) {
  const int b = threadIdx.x;
  __shared__ float ss[BATCH], nn[BATCH];
  float n = 0.f;
#pragma unroll
  for (int l = 0; l < 5; ++l) n += negpart[b * 5 + l];
  ss[b] = scoreb[b];
  nn[b] = n * 0.2f;
  __syncthreads();
  for (int st = 64; st; st >>= 1) {
    if (b < st) { ss[b] += ss[b + st]; nn[b] += nn[b + st]; }
    __syncthreads();
  }
  if (b == 0) {
    out[0] = (ss[0] + nn[0]) / (float)BATCH;
    out[1] = ss[0] / (float)BATCH;
    out[2] = nn[0] / (float)BATCH;
  }
}

// ================================ launcher ======================================
extern "C" void kernel_launch(void* const* d_in, const int* in_sizes, int n_in,
                              void* d_out, int out_size, void* d_ws, size_t ws_size,
                              hipStream_t stream) {
  const int*   pos_u = (const int*)d_in[0];
  const int*   pos_v = (const int*)d_in[1];
  const int*   neg_v = (const int*)d_in[2];
  const float* time  = (const float*)d_in[3];
  const float* m1    = (const float*)d_in[4];
  const float* b1    = (const float*)d_in[5];
  const float* W2    = (const float*)d_in[6];
  const float* b2    = (const float*)d_in[7];
  const float* W4    = (const float*)d_in[8];
  const float* b4    = (const float*)d_in[9];
  const float* U     = (const float*)d_in[10];
  const float* V     = (const float*)d_in[11];
  const float* T     = (const float*)d_in[12];
  const float* Bm    = (const float*)d_in[13];

  // workspace layout (floats); total ~18.5 MB
  float* ws      = (float*)d_ws;
  float* tv      = ws;                                  // 128*256
  float* tvB     = tv + BATCH * D;                      // 128*256
  float* E       = tvB + BATCH * D;                     // 896*768
  float* h3p     = E + (size_t)ROWS * D3;               // 12*896*256
  float* h3      = h3p + (size_t)KB_BLOCKS * ROWS * D;  // 896*256
  float* W4t     = h3 + (size_t)ROWS * D;               // 768*256
  float* use_    = W4t + (size_t)D3 * D;                // 896*768
  float* scoreb  = use_ + (size_t)ROWS * D3;            // 128
  float* negpart = scoreb + BATCH;                      // 640

  k_time<<<BATCH, 256, 0, stream>>>(time, m1, b1, W2, b2, Bm, tv, tvB);
  k_gather<<<dim3(BATCH, NSAMP), 256, 0, stream>>>(pos_u, pos_v, neg_v, U, V, E);
  k_contract<<<dim3(16, KB_BLOCKS), 256, SMEM_CONTRACT, stream>>>(T, tv, E, h3p);
  k_transpose<<<D3, 256, 0, stream>>>(W4, W4t);
  k_reduce<<<ROWS, 256, 0, stream>>>(h3p, tvB, h3);
  k_dense4<<<dim3(56, 6), 256, 0, stream>>>(h3, W4t, b4, use_);
  k_score<<<BATCH, 192, 0, stream>>>(use_, scoreb, negpart);
  k_final<<<1, 128, 0, stream>>>(scoreb, negpart, (float*)d_out);
}